// FFE_39410619908702
// MI455X (gfx1250) — compile-verified
//
#include <hip/hip_runtime.h>
#include <math.h>

typedef float v2f __attribute__((ext_vector_type(2)));
typedef float v8f __attribute__((ext_vector_type(8)));

#define N_ 8
#define C_ 64
#define H_ 128
#define W_ 128
#define HW_ (H_*W_)
#define GROUP_ 8
#define R_ 4
#define EPS_ 1e-5f

// ---- workspace layout (float offsets) ----
#define OFF_PATCH   0                            // N*C*9 shifted plane sums
#define OFF_FILT    (OFF_PATCH  + N_*C_*9)       // N*GROUP*9 softmaxed filter
#define OFF_ALOW    (OFF_FILT   + N_*GROUP_*9)   // N*C : 1+low_att
#define OFF_AHIGH   (OFF_ALOW   + N_*C_)         // N*C : 1+high_att
#define OFF_LOWPRE  (OFF_AHIGH  + N_*C_)         // N*C : low_pre
#define OFF_C1      (OFF_LOWPRE + N_*C_)         // R : folded bias for fusion fc1
#define OFF_W1      (OFF_C1     + R_)            // R*C : s1[j]*fu1_w[j][k]
#define OFF_W2      (OFF_W1     + R_*C_)         // C*R : s2[c]*fu2_w[c][j]
#define OFF_C2      (OFF_W2     + C_*R_)         // C : folded bias for fusion fc2

// ============================================================================
// Kernel 1: per (n,c) plane, compute 9 reflect-shifted sums in a single pass.
// Reflection pad=1 => shifted sum = weighted plane sum; row/col weight in {0,1,2}.
// ============================================================================
__global__ __launch_bounds__(256) void k1_patchsums(const float* __restrict__ x,
                                                    float* __restrict__ ws) {
  const int plane = blockIdx.x;                 // n*C + c
  const float* p = x + (size_t)plane * HW_;
  float acc[9];
#pragma unroll
  for (int k = 0; k < 9; ++k) acc[k] = 0.f;
  for (int idx = threadIdx.x; idx < HW_; idx += 256) {
    const int y  = idx >> 7;
    const int xw = idx & 127;
    const float v = p[idx];
    float wr[3], wc[3];
    wr[0] = 1.f + (y  == 1   ? 1.f : 0.f) - (y  == 127 ? 1.f : 0.f); // di=-1
    wr[1] = 1.f;
    wr[2] = 1.f + (y  == 126 ? 1.f : 0.f) - (y  == 0   ? 1.f : 0.f); // di=+1
    wc[0] = 1.f + (xw == 1   ? 1.f : 0.f) - (xw == 127 ? 1.f : 0.f);
    wc[1] = 1.f;
    wc[2] = 1.f + (xw == 126 ? 1.f : 0.f) - (xw == 0   ? 1.f : 0.f);
#pragma unroll
    for (int i = 0; i < 3; ++i)
#pragma unroll
      for (int j = 0; j < 3; ++j)
        acc[i*3+j] = fmaf(wr[i]*wc[j], v, acc[i*3+j]);
  }
  __shared__ float red[256][9];
#pragma unroll
  for (int k = 0; k < 9; ++k) red[threadIdx.x][k] = acc[k];
  __syncthreads();
  for (int s = 128; s > 0; s >>= 1) {
    if ((int)threadIdx.x < s)
#pragma unroll
      for (int k = 0; k < 9; ++k) red[threadIdx.x][k] += red[threadIdx.x + s][k];
    __syncthreads();
  }
  if (threadIdx.x == 0)
#pragma unroll
    for (int k = 0; k < 9; ++k) ws[OFF_PATCH + plane*9 + k] = red[0][k];
}

// ============================================================================
// Kernel 2 (single block): dynamic filter, SE branches, fusion BN folding.
// ============================================================================
__global__ __launch_bounds__(256) void k2_prep(
    float* __restrict__ ws,
    const float* __restrict__ conv_w, const float* __restrict__ bn_f,
    const float* __restrict__ fcl_w,  const float* __restrict__ fcl_b,
    const float* __restrict__ bn_l1,
    const float* __restrict__ fc0_w,  const float* __restrict__ fc0_b,
    const float* __restrict__ bn_l2,
    const float* __restrict__ fch_w,  const float* __restrict__ fch_b,
    const float* __restrict__ bn_h1,
    const float* __restrict__ fc1_w,  const float* __restrict__ fc1_b,
    const float* __restrict__ bn_h2,
    const float* __restrict__ fu1_w,  const float* __restrict__ fu1_b,
    const float* __restrict__ bn_fu1,
    const float* __restrict__ fu2_w,  const float* __restrict__ fu2_b,
    const float* __restrict__ bn_fu2) {
  const int t = threadIdx.x;
  __shared__ float s_gap[N_*C_];       // 512
  __shared__ float s_logit[N_*72];     // 576
  __shared__ float s_ml[N_*C_];
  __shared__ float s_mh[N_*C_];
  __shared__ float s_se[2*N_*R_];      // 64
  __shared__ float s_pre[2*N_*C_];     // 1024
  const float inv = 1.f / (float)HW_;

  // A: global average pool from center patch sum (p=4)
  for (int i = t; i < N_*C_; i += 256)
    s_gap[i] = ws[OFF_PATCH + i*9 + 4] * inv;
  __syncthreads();

  // B: filter logits = gap @ conv_w^T, then BN
  for (int i = t; i < N_*72; i += 256) {
    const int n = i / 72, o = i % 72;
    float s = 0.f;
    for (int c = 0; c < C_; ++c) s += s_gap[n*C_ + c] * conv_w[o*C_ + c];
    const float g = bn_f[0*72+o], b = bn_f[1*72+o];
    const float m = bn_f[2*72+o], v = bn_f[3*72+o];
    s_logit[i] = g * (s - m) * rsqrtf(v + EPS_) + b;
  }
  __syncthreads();

  // C: softmax over the 9 taps per (n, group)
  if (t < N_*GROUP_) {
    const int n = t / GROUP_, gi = t % GROUP_;
    const float* l = &s_logit[n*72 + gi*9];
    float mx = l[0];
    for (int k = 1; k < 9; ++k) mx = fmaxf(mx, l[k]);
    float e[9]; float sum = 0.f;
    for (int k = 0; k < 9; ++k) { e[k] = expf(l[k]-mx); sum += e[k]; }
    const float r = 1.f / sum;
    for (int k = 0; k < 9; ++k)
      ws[OFF_FILT + (n*GROUP_+gi)*9 + k] = e[k]*r;
  }
  __syncthreads();

  // D: mean(low_part), mean(high_part) per (n,c)
  for (int i = t; i < N_*C_; i += 256) {
    const int n = i / C_, c = i % C_;
    const float* f  = &ws[OFF_FILT + (n*GROUP_ + (c>>3))*9];
    const float* pssum = &ws[OFF_PATCH + i*9];
    float s = 0.f;
    for (int k = 0; k < 9; ++k) s += f[k]*pssum[k];
    s *= inv;
    s_ml[i] = s;
    s_mh[i] = s_gap[i] - s;
  }
  __syncthreads();

  // E: SE first FC + BN + ReLU (branch 0 = low, 1 = high)
  if (t < 2*N_*R_) {
    const int br = t / (N_*R_);
    const int n  = (t / R_) % N_;
    const int r  = t % R_;
    const float* w1  = br ? fch_w : fcl_w;
    const float* b1  = br ? fch_b : fcl_b;
    const float* bn1 = br ? bn_h1 : bn_l1;
    const float* inp = br ? s_mh  : s_ml;
    float s = b1[r];
    for (int c = 0; c < C_; ++c) s += inp[n*C_+c] * w1[r*C_+c];
    const float g = bn1[0*R_+r], bb = bn1[1*R_+r];
    const float m = bn1[2*R_+r], v  = bn1[3*R_+r];
    s = g * (s - m) * rsqrtf(v + EPS_) + bb;
    s_se[t] = s > 0.f ? s : 0.f;
  }
  __syncthreads();

  // F: SE second FC + BN -> pre-softmax logits
  for (int i = t; i < 2*N_*C_; i += 256) {
    const int br = i / (N_*C_);
    const int n  = (i / C_) % N_;
    const int c  = i % C_;
    const float* w2  = br ? fc1_w : fc0_w;
    const float* b2  = br ? fc1_b : fc0_b;
    const float* bn2 = br ? bn_h2 : bn_l2;
    const float* vv  = &s_se[br*N_*R_ + n*R_];
    float s = b2[c];
    for (int r = 0; r < R_; ++r) s += vv[r]*w2[c*R_+r];
    const float g = bn2[0*C_+c], bb = bn2[1*C_+c];
    const float m = bn2[2*C_+c], v  = bn2[3*C_+c];
    s = g * (s - m) * rsqrtf(v + EPS_) + bb;
    s_pre[i] = s;
    if (br == 0) ws[OFF_LOWPRE + n*C_ + c] = s;
  }
  __syncthreads();

  // G: channel softmax -> store (1 + attention)
  if (t < 2*N_) {
    const int br = t / N_, n = t % N_;
    const float* pr = &s_pre[br*N_*C_ + n*C_];
    float mx = pr[0];
    for (int c = 1; c < C_; ++c) mx = fmaxf(mx, pr[c]);
    float sum = 0.f;
    for (int c = 0; c < C_; ++c) sum += expf(pr[c]-mx);
    const float rs = 1.f / sum;
    float* dst = &ws[(br ? OFF_AHIGH : OFF_ALOW) + n*C_];
    for (int c = 0; c < C_; ++c) dst[c] = 1.f + expf(pr[c]-mx)*rs;
  }

  // H: fusion BN folding (independent of E..G for ws outputs)
  if (t < R_) {
    const float g = bn_fu1[0*R_+t], bb = bn_fu1[1*R_+t];
    const float m = bn_fu1[2*R_+t], v  = bn_fu1[3*R_+t];
    const float s1 = g * rsqrtf(v + EPS_);
    ws[OFF_C1 + t] = s1 * (fu1_b[t] - m) + bb;
  }
  {
    const int j = t >> 6;              // 0..3
    const int k = t & 63;              // 0..63
    const float g = bn_fu1[0*R_+j], v = bn_fu1[3*R_+j];
    const float s1 = g * rsqrtf(v + EPS_);
    ws[OFF_W1 + j*C_ + k] = s1 * fu1_w[j*C_ + k];
  }
  {
    const int c = t >> 2;              // 0..63
    const int j = t & 3;               // 0..3
    const float g = bn_fu2[0*C_+c], v = bn_fu2[3*C_+c];
    const float s2 = g * rsqrtf(v + EPS_);
    ws[OFF_W2 + c*R_ + j] = s2 * fu2_w[c*R_ + j];
    if (j == 0) {
      const float bb = bn_fu2[1*C_+c], m2 = bn_fu2[2*C_+c];
      ws[OFF_C2 + c] = s2 * (fu2_b[c] - m2) + bb;
    }
  }
}

// ============================================================================
// Kernel 3: main fused pass. Block = (n, h, 64-pixel segment), 128 threads.
// Phase 1: 9-tap dynamic low-pass, fea_low/fea_high into LDS.
// Phase 2: per-wave WMMA f32 16x16x4: v = relu(xa @ W1f^T + c1)  (K=64)
// Phase 3: per-pixel gate wei = sigmoid(W2·v + C2), final blend, store.
// ============================================================================
__global__ __launch_bounds__(128) void k3_main(const float* __restrict__ x,
                                               const float* __restrict__ ws,
                                               float* __restrict__ out) {
  const int wseg = blockIdx.x;           // 0..1
  const int h    = blockIdx.y;           // 0..127
  const int n    = blockIdx.z;           // 0..7
  const int t    = threadIdx.x;          // 0..127
  const int p    = t & 63;               // pixel in segment
  const int chalf= t >> 6;               // channel half
  const int w    = wseg*64 + p;

  __shared__ float s_fl[64][65];         // fea_low  [pixel][chan]
  __shared__ float s_fh[64][65];         // fea_high [pixel][chan]
  __shared__ float s_v [64][4];          // fusion hidden per pixel
  __shared__ float s_B [64][16];         // zero-padded B (K x Nout) for WMMA
  __shared__ float s_filt[GROUP_][9];

  // preload per-n filter + zero-padded B matrix
  if (t < GROUP_*9)
    s_filt[t/9][t%9] = ws[OFF_FILT + n*GROUP_*9 + t];
  for (int i = t; i < 64*16; i += 128) {
    const int k = i >> 4, col = i & 15;
    s_B[k][col] = (col < R_) ? ws[OFF_W1 + col*C_ + k] : 0.f;
  }
  __syncthreads();

  const int hm = (h == 0)      ? 1        : (h - 1);
  const int hp = (h == H_ - 1) ? (H_ - 2) : (h + 1);
  const int wm = (w == 0)      ? 1        : (w - 1);
  const int wp = (w == W_ - 1) ? (W_ - 2) : (w + 1);
  const size_t base_n = (size_t)n * C_ * HW_;

  // ---- phase 1: low-pass, fea_low / fea_high ----
  for (int j2 = 0; j2 < 32; ++j2) {
    const int c = chalf*32 + j2;
    const float* xp = x + base_n + (size_t)c * HW_;
    const float* r0 = xp + (size_t)hm * W_;
    const float* r1 = xp + (size_t)h  * W_;
    const float* r2 = xp + (size_t)hp * W_;
    const float* f  = s_filt[c >> 3];
    float low = f[0]*r0[wm] + f[1]*r0[w] + f[2]*r0[wp]
              + f[3]*r1[wm] + f[4]*r1[w] + f[5]*r1[wp]
              + f[6]*r2[wm] + f[7]*r2[w] + f[8]*r2[wp];
    const float xc = r1[w];
    const float ah = ws[OFF_AHIGH  + n*C_ + c];
    const float al = ws[OFF_ALOW   + n*C_ + c];
    const float lp = ws[OFF_LOWPRE + n*C_ + c];
    s_fh[p][c] = (xc - low) * ah;            // high*(1+att_h)
    s_fl[p][c] = low * al + lp;              // low*(1+att_l)+low_pre
  }
  __syncthreads();

  // ---- phase 2: WMMA 16x16x4 f32, K=64 accumulation (all 4 waves active) ----
  {
    const int lane = t & 31;
    const int wv   = t >> 5;               // wave id 0..3 -> 16-pixel tile
    const int row  = wv*16 + (lane & 15);  // A-matrix M (pixel)
    const int koff = (lane < 16) ? 0 : 2;  // A/B fragment K half
    const int col  = lane & 15;            // B-matrix N
    v8f acc = {};
#pragma unroll
    for (int k0 = 0; k0 < C_; k0 += 4) {
      v2f a, b;
      a.x = s_fl[row][k0+koff]   + s_fh[row][k0+koff];     // xa = fea_low+fea_high
      a.y = s_fl[row][k0+koff+1] + s_fh[row][k0+koff+1];
      b.x = s_B[k0+koff][col];
      b.y = s_B[k0+koff+1][col];
      acc = __builtin_amdgcn_wmma_f32_16x16x4_f32(false, a, false, b,
                                                  (short)0, acc, false, false);
    }
    if (col < R_) {
      const float c1 = ws[OFF_C1 + col];
      const int mbase = wv*16 + ((lane < 16) ? 0 : 8);
#pragma unroll
      for (int r = 0; r < 8; ++r) {
        const float vv = acc[r] + c1;       // BN already folded into W1f
        s_v[mbase + r][col] = vv > 0.f ? vv : 0.f;
      }
    }
  }
  __syncthreads();

  // ---- phase 3: gate + blend + store ----
  const float v0 = s_v[p][0], v1 = s_v[p][1], v2 = s_v[p][2], v3 = s_v[p][3];
  for (int j2 = 0; j2 < 32; ++j2) {
    const int c = chalf*32 + j2;
    const float* w2 = ws + OFF_W2 + c*R_;
    const float lg  = ws[OFF_C2 + c] + w2[0]*v0 + w2[1]*v1 + w2[2]*v2 + w2[3]*v3;
    const float wei = 1.f / (1.f + expf(-lg));
    const float fl = s_fl[p][c], fh = s_fh[p][c];
    // fea_low*wei + fea_high*(1-wei) + fea_low + fea_high
    out[base_n + (size_t)c*HW_ + (size_t)h*W_ + w] = fl*(1.f + wei) + fh*(2.f - wei);
  }
}

// ============================================================================
extern "C" void kernel_launch(void* const* d_in, const int* in_sizes, int n_in,
                              void* d_out, int out_size, void* d_ws, size_t ws_size,
                              hipStream_t stream) {
  (void)in_sizes; (void)n_in; (void)out_size; (void)ws_size;
  const float* x      = (const float*)d_in[0];
  const float* conv_w = (const float*)d_in[1];
  const float* bn_f   = (const float*)d_in[2];
  const float* fcl_w  = (const float*)d_in[3];
  const float* fcl_b  = (const float*)d_in[4];
  const float* bn_l1  = (const float*)d_in[5];
  const float* fc0_w  = (const float*)d_in[6];
  const float* fc0_b  = (const float*)d_in[7];
  const float* bn_l2  = (const float*)d_in[8];
  const float* fch_w  = (const float*)d_in[9];
  const float* fch_b  = (const float*)d_in[10];
  const float* bn_h1  = (const float*)d_in[11];
  const float* fc1_w  = (const float*)d_in[12];
  const float* fc1_b  = (const float*)d_in[13];
  const float* bn_h2  = (const float*)d_in[14];
  const float* fu1_w  = (const float*)d_in[15];
  const float* fu1_b  = (const float*)d_in[16];
  const float* bn_fu1 = (const float*)d_in[17];
  const float* fu2_w  = (const float*)d_in[18];
  const float* fu2_b  = (const float*)d_in[19];
  const float* bn_fu2 = (const float*)d_in[20];
  float* ws  = (float*)d_ws;
  float* out = (float*)d_out;

  k1_patchsums<<<dim3(N_*C_), dim3(256), 0, stream>>>(x, ws);
  k2_prep<<<dim3(1), dim3(256), 0, stream>>>(ws,
      conv_w, bn_f, fcl_w, fcl_b, bn_l1, fc0_w, fc0_b, bn_l2,
      fch_w, fch_b, bn_h1, fc1_w, fc1_b, bn_h2,
      fu1_w, fu1_b, bn_fu1, fu2_w, fu2_b, bn_fu2);
  k3_main<<<dim3(W_/64, H_, N_), dim3(128), 0, stream>>>(x, ws, out);
}